// Light_GCN_24446953849419
// MI455X (gfx1250) — compile-verified
//
#include <hip/hip_runtime.h>
#include <hip/hip_bf16.h>

// ---------------------------------------------------------------------------
// LightGCN on MI455X (gfx1250, wave32)
//   - 3x COO SpMM with L2-resident embedding buffers (3 x 38.4MB < 192MB L2)
//   - hardware f32 global atomics for the segment-sum scatter
//   - global_prefetch_b8 on the streamed edge lists (only HBM traffic)
//   - final batched dot-product scoring via V_WMMA_F32_16X16X4_F32 (diag of
//     U_tile x I_tile^T, accumulated over K=64 in 16 chunks of 4)
// ---------------------------------------------------------------------------

typedef __attribute__((ext_vector_type(2))) float v2f;
typedef __attribute__((ext_vector_type(8))) float v8f;

#define EMBED 64

// ---------------------------------------------------------------------------
// init: e_cur = acc = concat(user_emb, item_emb); e_next = 0   (float4 wide)
// ---------------------------------------------------------------------------
__global__ void lgcn_init_kernel(const float4* __restrict__ ue,
                                 const float4* __restrict__ ie,
                                 float4* __restrict__ e_cur,
                                 float4* __restrict__ acc,
                                 float4* __restrict__ e_next,
                                 int n_user_f4, int total_f4) {
    int i = blockIdx.x * blockDim.x + threadIdx.x;
    if (i >= total_f4) return;
    float4 v = (i < n_user_f4) ? ue[i] : ie[i - n_user_f4];
    e_cur[i]  = v;
    acc[i]    = v;
    e_next[i] = make_float4(0.f, 0.f, 0.f, 0.f);
}

// ---------------------------------------------------------------------------
// SpMM (COO, edge-parallel): e_out[r,:] += v * e_in[c,:]
//   One wave per group of 32 edges: coalesced index loads staged per lane,
//   then broadcast via ds_bpermute (__shfl); each lane owns dims
//   {2*lane, 2*lane+1} so the row gather is a coalesced 256B float2 load and
//   the scatter is two global_atomic_add_f32 (L2-resident destination).
// ---------------------------------------------------------------------------
__global__ void lgcn_spmm_kernel(const float* __restrict__ e_in,
                                 float* __restrict__ e_out,
                                 const float* __restrict__ vals,
                                 const int* __restrict__ rows,
                                 const int* __restrict__ cols,
                                 int nnz) {
    int wave   = (blockIdx.x * blockDim.x + threadIdx.x) >> 5;
    int lane   = threadIdx.x & 31;
    int nwaves = (gridDim.x * blockDim.x) >> 5;
    int stride = nwaves * 32;

    for (int base = wave * 32; base < nnz; base += stride) {
        int e = base + lane;
        int r = 0, c = 0;
        float v = 0.0f;
        if (e < nnz) {
            r = rows[e];
            c = cols[e];
            v = vals[e];
        }
        // Prefetch next edge block (edge lists are the only HBM streaming).
        int nb = base + stride + lane;
        if (nb < nnz) {
            __builtin_prefetch(&rows[nb], 0, 0);
            __builtin_prefetch(&cols[nb], 0, 0);
            __builtin_prefetch(&vals[nb], 0, 0);
        }
        int cnt = nnz - base;
        if (cnt > 32) cnt = 32;
        for (int j = 0; j < cnt; ++j) {
            int   rj = __shfl(r, j, 32);
            int   cj = __shfl(c, j, 32);
            float vj = __shfl(v, j, 32);
            float2 s = ((const float2*)(e_in + (size_t)cj * EMBED))[lane];
            float* dst = e_out + (size_t)rj * EMBED + lane * 2;
            unsafeAtomicAdd(dst,     vj * s.x);   // global_atomic_add_f32
            unsafeAtomicAdd(dst + 1, vj * s.y);
        }
    }
}

// ---------------------------------------------------------------------------
// post: acc += e_new; zero the retiring buffer for the next layer
// ---------------------------------------------------------------------------
__global__ void lgcn_post_kernel(float4* __restrict__ acc,
                                 const float4* __restrict__ e_new,
                                 float4* __restrict__ to_zero,
                                 int total_f4) {
    int i = blockIdx.x * blockDim.x + threadIdx.x;
    if (i >= total_f4) return;
    float4 a = acc[i];
    float4 e = e_new[i];
    a.x += e.x; a.y += e.y; a.z += e.z; a.w += e.w;
    acc[i] = a;
    to_zero[i] = make_float4(0.f, 0.f, 0.f, 0.f);
}

// ---------------------------------------------------------------------------
// score: one wave per tile of 16 batch entries.
//   D = U_tile (16x64) x I_tile^T (64x16) via 16 chained
//   V_WMMA_F32_16X16X4_F32 ops; score_b = 0.0625 * D[b,b].
//   A-matrix 16x4 f32 layout (ISA 7.12.2): lane L holds M = L%16,
//   VGPR0 -> K = (L<16 ? 0 : 2), VGPR1 -> K+1; B mirrors with N = L%16.
//   A and B slots are filled with the identical K mapping, so the diagonal
//   contracts u_b . i_b over all 64 features exactly once.
//   Diagonal extraction uses a constant-index cndmask select chain (no
//   dynamic vector indexing -> no scratch spill).
// ---------------------------------------------------------------------------
__global__ void lgcn_score_kernel(const float* __restrict__ acc,
                                  const int* __restrict__ user,
                                  const int* __restrict__ item,
                                  float* __restrict__ out,
                                  int n_user, int batch) {
    int wave = (blockIdx.x * blockDim.x + threadIdx.x) >> 5;
    int lane = threadIdx.x & 31;
    int b0   = wave * 16;
    if (b0 >= batch) return;          // wave-uniform exit; EXEC all-1s below

    int m  = lane & 15;               // row within tile (both A and B)
    int hi = lane >> 4;               // half-wave selects K pairs {0,1}/{2,3}

    int u  = user[b0 + m];
    int it = item[b0 + m];
    const float* urow = acc + (size_t)u * EMBED;
    const float* irow = acc + (size_t)(n_user + it) * EMBED;

    v8f c = {0.f, 0.f, 0.f, 0.f, 0.f, 0.f, 0.f, 0.f};
#pragma unroll
    for (int kk = 0; kk < EMBED; kk += 4) {
        int base = kk + hi * 2;
        v2f a, b;
        a.x = urow[base];
        a.y = urow[base + 1];
        b.x = irow[base];
        b.y = irow[base + 1];
        // 8 args: (neg_a, A, neg_b, B, c_mod, C, reuse_a, reuse_b)
        c = __builtin_amdgcn_wmma_f32_16x16x4_f32(
                false, a, false, b, (short)0, c, false, false);
    }

    // C/D 16x16 f32 layout: vgpr r, lanes 0-15 -> M=r, lanes 16-31 -> M=r+8;
    // N = lane%16.  Diagonal (b,b): b<8 -> lane b, vgpr b;
    //                              b>=8 -> lane b+16, vgpr b-8.
    bool active = (lane < 8) | (lane >= 24);
    int  sel    = (lane < 8) ? lane : (lane - 24);   // accumulator VGPR index
    int  bidx   = (lane < 8) ? lane : (lane - 16);   // batch element in tile

    // Constant-index extracts + cndmask chain (stays in VGPRs, no scratch).
    float sc = c[0];
    sc = (sel == 1) ? c[1] : sc;
    sc = (sel == 2) ? c[2] : sc;
    sc = (sel == 3) ? c[3] : sc;
    sc = (sel == 4) ? c[4] : sc;
    sc = (sel == 5) ? c[5] : sc;
    sc = (sel == 6) ? c[6] : sc;
    sc = (sel == 7) ? c[7] : sc;

    const float scale = 0.0625f;      // (1/(L+1))^2 = (1/4)^2
    if (active) out[b0 + bidx] = scale * sc;
}

// ---------------------------------------------------------------------------
// host launcher (graph-capture safe: only kernel launches on `stream`)
// inputs: user_emb, item_emb, adj_vals, adj_rows, adj_cols, user, item
// ---------------------------------------------------------------------------
extern "C" void kernel_launch(void* const* d_in, const int* in_sizes, int n_in,
                              void* d_out, int out_size, void* d_ws, size_t ws_size,
                              hipStream_t stream) {
    const float* user_emb = (const float*)d_in[0];
    const float* item_emb = (const float*)d_in[1];
    const float* vals     = (const float*)d_in[2];
    const int*   rows     = (const int*)d_in[3];
    const int*   cols     = (const int*)d_in[4];
    const int*   user     = (const int*)d_in[5];
    const int*   item     = (const int*)d_in[6];
    float*       out      = (float*)d_out;

    const int n_user = in_sizes[0] / EMBED;
    const int n_item = in_sizes[1] / EMBED;
    const int nnz    = in_sizes[2];
    const int batch  = in_sizes[5];
    const int N      = n_user + n_item;

    // Workspace: 3 x (N*64 f32) buffers = ~115 MB (fits 192 MB L2 resident)
    const size_t buf_bytes = (size_t)N * EMBED * sizeof(float);
    float* buf0 = (float*)d_ws;
    float* buf1 = (float*)((char*)d_ws + buf_bytes);
    float* accb = (float*)((char*)d_ws + 2 * buf_bytes);

    const int total_f4  = N * (EMBED / 4);
    const int n_user_f4 = n_user * (EMBED / 4);
    const int eb        = (total_f4 + 255) / 256;

    lgcn_init_kernel<<<eb, 256, 0, stream>>>(
        (const float4*)user_emb, (const float4*)item_emb,
        (float4*)buf0, (float4*)accb, (float4*)buf1, n_user_f4, total_f4);

    float* cur = buf0;
    float* nxt = buf1;
    for (int l = 0; l < 3; ++l) {
        lgcn_spmm_kernel<<<4096, 256, 0, stream>>>(cur, nxt, vals, rows, cols, nnz);
        lgcn_post_kernel<<<eb, 256, 0, stream>>>(
            (float4*)accb, (const float4*)nxt, (float4*)cur, total_f4);
        float* t = cur; cur = nxt; nxt = t;
    }

    const int tiles  = (batch + 15) / 16;           // 512 waves
    const int blocks = (tiles + 7) / 8;             // 8 waves per 256-thread block
    lgcn_score_kernel<<<blocks, 256, 0, stream>>>(accb, user, item, out,
                                                  n_user, batch);
}